// StyledConvLayer_41351945125948
// MI455X (gfx1250) — compile-verified
//
#include <hip/hip_runtime.h>
#include <hip/hip_bf16.h>
#include <math.h>
#include <stdint.h>

// Problem constants (match reference)
#define BATCH   16
#define IN_CH   128
#define OUT_CH  256
#define STYLE_D 512
#define KS      3
#define L_LEN   16384
#define IK      (IN_CH * KS)       // 384 contraction length, ordered r = k*128 + i

// Main-kernel tiling
#define NT        64               // L columns per workgroup
#define NTILES    (NT / 16)        // 4 WMMA N-tiles
#define KSTEPS1   (IK / 4)         // 96 K-steps of 4 for GEMM1
#define KSTEPS2   (IN_CH / 4)      // 32 K-steps of 4 for GEMM2
#define XS_COLS   (NT + 2)         // 66 (halo for KS=3, pad=1)
#define XS_STRIDE (IN_CH + 2)      // 130 floats: padded to break LDS bank conflicts
#define GS_OFF    (XS_COLS * XS_STRIDE)  // xs occupies [0, 8580)

typedef __attribute__((ext_vector_type(2))) float v2f;
typedef __attribute__((ext_vector_type(8))) float v8f;

#define SCALE_MOD  0.044194173824159216f   // 1/sqrt(512)
#define SCALE_CONV 0.029462782549439483f   // 1/sqrt(128*9)
#define SQRT_HALF  0.70710678118654752f
#define LOG2E      1.44269504088896340736f

// ---------------------------------------------------------------------------
// Prep kernel 1: s[b,i] = style[b,:] . (mod_weight[i,:]*scale_mod) + mod_bias[i]
__global__ __launch_bounds__(256) void k_mod(const float* __restrict__ style,
                                             const float* __restrict__ mw,
                                             const float* __restrict__ mb,
                                             float* __restrict__ s) {
  int t = blockIdx.x * 256 + threadIdx.x;       // 2048 = B*IN_CH
  int b = t >> 7, i = t & 127;
  float sum = mb[i];
  for (int j = 0; j < STYLE_D; ++j)
    sum += style[b * STYLE_D + j] * mw[i * STYLE_D + j] * SCALE_MOD;
  s[t] = sum;
}

// Prep kernel 2: demod[b,o] = rsqrt(sum_{i,k} (scale*cw[o,i,k]*s[b,i])^2 + 1e-8)
__global__ __launch_bounds__(256) void k_demod(const float* __restrict__ cw,
                                               const float* __restrict__ s,
                                               float* __restrict__ demod) {
  int t = blockIdx.x * 256 + threadIdx.x;       // 4096 = B*OUT_CH
  int b = t >> 8, o = t & 255;
  float sum = 0.f;
  for (int i = 0; i < IN_CH; ++i) {
    float si = s[b * IN_CH + i] * SCALE_CONV;
    for (int k = 0; k < KS; ++k) {
      float v = cw[(o * IN_CH + i) * KS + k] * si;
      sum += v * v;
    }
  }
  demod[t] = rsqrtf(sum + 1e-8f);
}

// Prep kernel 3: scatter modulated+demodulated weight into WMMA A-fragment
// layout: wA[((b*16+mt)*96+ks)*32 + lane] is a float2 (two K values per lane).
// Contraction index r = k*128 + i ; K-step ks = k*32 + (i>>2).
// A-frag lane j holds M=(j&15), K offsets d = 2*(j>>4)+{0,1}.
__global__ __launch_bounds__(256) void k_wa(const float* __restrict__ cw,
                                            const float* __restrict__ s,
                                            const float* __restrict__ demod,
                                            float* __restrict__ wA) {
  int t = blockIdx.x * 256 + threadIdx.x;       // 1,572,864 = B*O*I*K
  int k = t % 3;
  int i = (t / 3) & 127;
  int o = (t / (IN_CH * KS)) & 255;
  int b = t / (OUT_CH * IN_CH * KS);
  float val = SCALE_CONV * cw[(o * IN_CH + i) * KS + k]
            * s[b * IN_CH + i] * demod[b * OUT_CH + o];
  int mt = o >> 4;
  int ks = k * 32 + (i >> 2);
  int ln = (o & 15) + (((i >> 1) & 1) << 4);
  int v  = i & 1;
  wA[((((size_t)b * 16 + mt) * KSTEPS1 + ks) * 32 + ln) * 2 + v] = val;
}

// Prep kernel 4: weight-normed 1x1 conv weight, pre-swizzled A-frag layout
// w1[o,c] = out_g[o]*out_v[o,c]/||out_v[o,:]||  ->  w1A[(mt*32+ks)*32+lane] float2
__global__ __launch_bounds__(256) void k_w1(const float* __restrict__ out_v,
                                            const float* __restrict__ out_g,
                                            float* __restrict__ w1A) {
  int t = blockIdx.x * 256 + threadIdx.x;       // 16384 = 128*128
  int o = t >> 7, c = t & 127;
  float nrm = 0.f;
  for (int cc = 0; cc < 128; ++cc) {
    float vv = out_v[o * 128 + cc];
    nrm += vv * vv;
  }
  float val = out_g[o] * out_v[o * 128 + c] * rsqrtf(nrm);
  int mt = o >> 4, ks = c >> 2;
  int ln = (o & 15) + (((c >> 1) & 1) << 4);
  int v  = c & 1;
  w1A[(((size_t)mt * KSTEPS2 + ks) * 32 + ln) * 2 + v] = val;
}

// ---------------------------------------------------------------------------
// Fused main kernel: conv GEMM (WMMA) -> gate -> 1x1 GEMM (WMMA) -> residual.
// One workgroup = one sample b, one 64-column L tile, all 256 out channels.
// 8 wave32s; wave w owns conv channels [16w,16w+16) and [128+16w,128+16w+16).
__global__ __launch_bounds__(256)
void styled_main(const float* __restrict__ x, const float* __restrict__ wA,
                 const float* __restrict__ w1A, const float* __restrict__ out_bias,
                 float* __restrict__ out) {
  extern __shared__ float smem[];
  float* xs = smem;            // [c in 0..65][i in 0..127], stride 130 (col-major)
  float* gs = smem + GS_OFF;   // [n in 0..63][c in 0..127], stride 130 (col-major)

  const int b    = blockIdx.y;
  const int l0   = blockIdx.x * NT;
  const int t    = threadIdx.x;
  const int wave = t >> 5;
  const int lane = t & 31;
  const int nrow = lane & 15;          // N index within a 16-tile
  const int ioff = (lane >> 4) << 1;   // K-pair offset: lanes 0-15 -> 0, 16-31 -> 2

  // ---- stage x tile into LDS (col-major, transposed) ----
  const float* xb = x + (size_t)b * IN_CH * L_LEN;

  // Interior columns c in [1, 65) are always in-range: CDNA5 async global->LDS.
  // LDS byte address = low 32 bits of the generic pointer (LDS aperture).
  {
    unsigned lds_base = (unsigned)(uintptr_t)xs;
    for (int e = t; e < IN_CH * NT; e += 256) {
      int c = 1 + (e & (NT - 1));           // consecutive lanes -> consecutive l
      int i = e >> 6;
      unsigned lds_off = lds_base + (unsigned)((c * XS_STRIDE + i) << 2);
      const float* gp  = xb + (size_t)i * L_LEN + (l0 + c - 1);
      asm volatile("global_load_async_to_lds_b32 %0, %1, off"
                   :: "v"(lds_off), "v"(gp)
                   : "memory");
    }
  }
  // Halo columns c=0 and c=65 need zero-fill at the sequence boundary.
  {
    int i  = t & 127;
    int c  = (t < 128) ? 0 : (XS_COLS - 1);
    int gl = l0 + c - 1;
    float v = (gl >= 0 && gl < L_LEN) ? xb[(size_t)i * L_LEN + gl] : 0.0f;
    xs[c * XS_STRIDE + i] = v;
  }
#if __has_builtin(__builtin_amdgcn_s_wait_asynccnt)
  __builtin_amdgcn_s_wait_asynccnt(0);
#else
  asm volatile("s_wait_asynccnt 0" ::: "memory");
#endif
  __syncthreads();

  // ---- GEMM1: out[o, n] = sum_r wA[o, r] * x-patches[r, n], r = k*128+i ----
  v8f acc0[NTILES] = {};   // channels 16w .. 16w+15       (xa half)
  v8f acc1[NTILES] = {};   // channels 128+16w .. +15      (xb half)
  const v2f* pA0 = (const v2f*)wA + ((size_t)(b * 16 + wave)     * KSTEPS1) * 32 + lane;
  const v2f* pA1 = (const v2f*)wA + ((size_t)(b * 16 + wave + 8) * KSTEPS1) * 32 + lane;

  for (int ks = 0; ks < KSTEPS1; ++ks) {
    v2f a0 = pA0[ks * 32];
    v2f a1 = pA1[ks * 32];
    if (ks + 8 < KSTEPS1) {
      __builtin_prefetch((const void*)(pA0 + (ks + 8) * 32), 0, 3);
      __builtin_prefetch((const void*)(pA1 + (ks + 8) * 32), 0, 3);
    }
    int k  = ks >> 5;            // conv tap 0..2
    int i0 = (ks & 31) << 2;     // base input channel of this K-step
    const float* bbase = &xs[(nrow + k) * XS_STRIDE + i0 + ioff];
#pragma unroll
    for (int nt = 0; nt < NTILES; ++nt) {
      v2f bf = *(const v2f*)(bbase + nt * 16 * XS_STRIDE);
      acc0[nt] = __builtin_amdgcn_wmma_f32_16x16x4_f32(
          false, a0, false, bf, (short)0, acc0[nt], false, false);
      acc1[nt] = __builtin_amdgcn_wmma_f32_16x16x4_f32(
          false, a1, false, bf, (short)0, acc1[nt], false, false);
    }
  }

  // ---- gate: g = tanh(xa) * sigmoid(xb); store to LDS col-major ----
  // Use gfx1250 hardware transcendentals where available (v_tanh_f32,
  // v_exp_f32, v_rcp_f32) instead of the libm expansions.
#pragma unroll
  for (int nt = 0; nt < NTILES; ++nt) {
    int n = nt * 16 + nrow;
#pragma unroll
    for (int v = 0; v < 8; ++v) {
      int row = wave * 16 + v + ((lane >> 4) << 3);   // C layout: M = v (+8 for hi lanes)
      float ga = acc0[nt][v];
      float gb = acc1[nt][v];
#if __has_builtin(__builtin_amdgcn_tanhf)
      float th = __builtin_amdgcn_tanhf(ga);
#else
      float th = tanhf(ga);
#endif
#if __has_builtin(__builtin_amdgcn_exp2f) && __has_builtin(__builtin_amdgcn_rcpf)
      float sg = __builtin_amdgcn_rcpf(1.0f + __builtin_amdgcn_exp2f(gb * -LOG2E));
#else
      float sg = 1.0f / (1.0f + __expf(-gb));
#endif
      gs[n * XS_STRIDE + row] = th * sg;
    }
  }
  __syncthreads();

  // ---- GEMM2: y[o, n] = sum_c w1[o, c] * g[c, n], wave w -> rows 16w..16w+15 ----
  v8f acc2[NTILES] = {};
  const v2f* pW1 = (const v2f*)w1A + ((size_t)wave * KSTEPS2) * 32 + lane;
  for (int ks = 0; ks < KSTEPS2; ++ks) {
    v2f a  = pW1[ks * 32];
    int c0 = (ks << 2) + ioff;
#pragma unroll
    for (int nt = 0; nt < NTILES; ++nt) {
      v2f bf = *(const v2f*)&gs[(nt * 16 + nrow) * XS_STRIDE + c0];
      acc2[nt] = __builtin_amdgcn_wmma_f32_16x16x4_f32(
          false, a, false, bf, (short)0, acc2[nt], false, false);
    }
  }

  // ---- epilogue: + bias, + residual (from xs tile), * sqrt(0.5) ----
  float* ob = out + (size_t)b * IN_CH * L_LEN + l0;
#pragma unroll
  for (int nt = 0; nt < NTILES; ++nt) {
    int n = nt * 16 + nrow;
#pragma unroll
    for (int v = 0; v < 8; ++v) {
      int row  = wave * 16 + v + ((lane >> 4) << 3);
      float y  = acc2[nt][v] + out_bias[row];
      float xr = xs[(n + 1) * XS_STRIDE + row];   // x[b,row,l0+n] (halo offset +1)
      ob[(size_t)row * L_LEN + n] = (y + xr) * SQRT_HALF;
    }
  }
}

// ---------------------------------------------------------------------------
extern "C" void kernel_launch(void* const* d_in, const int* in_sizes, int n_in,
                              void* d_out, int out_size, void* d_ws, size_t ws_size,
                              hipStream_t stream) {
  const float* x     = (const float*)d_in[0];
  const float* style = (const float*)d_in[1];
  const float* cw    = (const float*)d_in[2];
  const float* mw    = (const float*)d_in[3];
  const float* mb    = (const float*)d_in[4];
  const float* ov    = (const float*)d_in[5];
  const float* og    = (const float*)d_in[6];
  const float* obias = (const float*)d_in[7];
  float* out = (float*)d_out;

  float* ws    = (float*)d_ws;
  float* s     = ws;                 // 2,048 floats
  float* demod = ws + 2048;          // 4,096 floats
  float* w1A   = ws + 6144;          // 16,384 floats
  float* wA    = ws + 22528;         // 1,572,864 floats (~6.4 MB total)

  k_mod  <<<   8, 256, 0, stream>>>(style, mw, mb, s);
  k_demod<<<  16, 256, 0, stream>>>(cw, s, demod);
  k_wa   <<<6144, 256, 0, stream>>>(cw, s, demod, wA);
  k_w1   <<<  64, 256, 0, stream>>>(ov, og, w1A);

  dim3 grid(L_LEN / NT, BATCH);                       // (256, 16)
  size_t smem = (size_t)(GS_OFF + NT * XS_STRIDE) * sizeof(float);  // 67,600 B
  styled_main<<<grid, 256, smem, stream>>>(x, wA, w1A, obias, out);
}